// LocalAttnTransformer_45913200394621
// MI455X (gfx1250) — compile-verified
//
#include <hip/hip_runtime.h>
#include <hip/hip_bf16.h>
#include <math.h>
#include <stdint.h>

typedef __attribute__((ext_vector_type(2))) float v2f;
typedef __attribute__((ext_vector_type(8))) float v8f;

#define MROWS 8192          // 2 * 64 * 64
#define CCH   256
#define KS2   49
#define NHEAD 8
#define HDIM  32

// ---------------------------------------------------------------------------
// CDNA5 async global->LDS copy (128b per lane), tracked by ASYNCcnt.
// Writes LDS directly, no VGPR round trip (ISA 15.18.3 op 98).
// ---------------------------------------------------------------------------
__device__ __forceinline__ void async_copy_b128(void* lds, const void* gptr) {
  uint32_t l = (uint32_t)(uintptr_t)lds;   // flat LDS addr: low 32 bits = LDS offset
  asm volatile("global_load_async_to_lds_b128 %0, %1, off"
               :: "v"(l), "v"(gptr) : "memory");
}

// ---------------------------------------------------------------------------
// Layout transpose: NCHW [2,256,64,64] <-> NHWC rows [8192, 256]
// ---------------------------------------------------------------------------
__global__ __launch_bounds__(256) void nchw_to_nhwc_kernel(const float* __restrict__ in,
                                                           float* __restrict__ out) {
  int idx = blockIdx.x * 256 + threadIdx.x;   // over 2*256*4096, NCHW order
  int p = idx & 4095;
  int c = (idx >> 12) & 255;
  int n = idx >> 20;
  int m = (n << 12) + p;
  out[(size_t)m * CCH + c] = in[idx];
}

__global__ __launch_bounds__(256) void nhwc_to_nchw_kernel(const float* __restrict__ in,
                                                           float* __restrict__ out) {
  int idx = blockIdx.x * 256 + threadIdx.x;   // over NCHW order of output
  int p = idx & 4095;
  int c = (idx >> 12) & 255;
  int n = idx >> 20;
  int m = (n << 12) + p;
  out[idx] = in[(size_t)m * CCH + c];
}

// ---------------------------------------------------------------------------
// WMMA fp32 GEMM:  out[M,N] = A[M,K] * W[N,K]^T + bias (+relu) (+residual)
// Block: 256 threads = 8 waves (2x4); block tile 64x128, wave tile 32x32.
// A and B staged in LDS via double-buffered GLOBAL_LOAD_ASYNC_TO_LDS_B128:
// 6 async b128 copies per thread per 32-wide K chunk -> s_wait_asynccnt 6/0.
// grid = (N/128, M/64). K multiple of 32.
// ---------------------------------------------------------------------------
__global__ __launch_bounds__(256) void wmma_gemm_kernel(
    const float* __restrict__ A, const float* __restrict__ W,
    const float* __restrict__ bias, const float* __restrict__ residual,
    float* __restrict__ out, int M, int N, int K, int relu) {
  __shared__ float As[2][64][36];    // stride 36: 16B-aligned rows, conflict-free b64 reads
  __shared__ float Bs[2][128][36];   // total 55,296 B LDS (of 320 KB/WGP)

  const int tid  = threadIdx.x;
  const int wave = tid >> 5;
  const int lane = tid & 31;
  const int l16  = lane & 15;           // A-frag row / B-frag col
  const int kb   = (lane >> 4) << 1;    // K sub-offset per lane half (0 or 2)
  const int wr   = (wave >> 2) << 5;    // wave row offset in block tile: 0,32
  const int wc   = (wave & 3) << 5;     // wave col offset: 0,32,64,96

  const int blockM = blockIdx.y << 6;
  const int blockN = blockIdx.x << 7;

  const int lrow = tid >> 3;            // 0..31: coop-load row
  const int lkof = (tid & 7) << 2;      // 0..28: coop-load k offset (float4)

  v8f c00 = {}, c01 = {}, c10 = {}, c11 = {};

  const int nch = K >> 5;

  // prologue: stage chunk 0 into buffer 0
#pragma unroll
  for (int rep = 0; rep < 2; ++rep)
    async_copy_b128(&As[0][rep * 32 + lrow][lkof],
                    A + (size_t)(blockM + rep * 32 + lrow) * K + lkof);
#pragma unroll
  for (int rep = 0; rep < 4; ++rep)
    async_copy_b128(&Bs[0][rep * 32 + lrow][lkof],
                    W + (size_t)(blockN + rep * 32 + lrow) * K + lkof);

  for (int ic = 0; ic < nch; ++ic) {
    const int bsel = ic & 1;
    if (ic + 1 < nch) {
      // prefetch next chunk into the other buffer, then wait for current 6
      const int kc = (ic + 1) << 5;
      const int nb = bsel ^ 1;
#pragma unroll
      for (int rep = 0; rep < 2; ++rep)
        async_copy_b128(&As[nb][rep * 32 + lrow][lkof],
                        A + (size_t)(blockM + rep * 32 + lrow) * K + kc + lkof);
#pragma unroll
      for (int rep = 0; rep < 4; ++rep)
        async_copy_b128(&Bs[nb][rep * 32 + lrow][lkof],
                        W + (size_t)(blockN + rep * 32 + lrow) * K + kc + lkof);
      asm volatile("s_wait_asynccnt 0x6" ::: "memory");  // oldest 6 (= chunk ic) done
    } else {
      asm volatile("s_wait_asynccnt 0x0" ::: "memory");
    }
    __syncthreads();

#pragma unroll
    for (int kl = 0; kl < 32; kl += 4) {
      v2f a0 = *(const v2f*)&As[bsel][wr + l16][kl + kb];
      v2f a1 = *(const v2f*)&As[bsel][wr + l16 + 16][kl + kb];
      v2f b0 = *(const v2f*)&Bs[bsel][wc + l16][kl + kb];
      v2f b1 = *(const v2f*)&Bs[bsel][wc + l16 + 16][kl + kb];
      c00 = __builtin_amdgcn_wmma_f32_16x16x4_f32(false, a0, false, b0, (short)0, c00, false, false);
      c01 = __builtin_amdgcn_wmma_f32_16x16x4_f32(false, a0, false, b1, (short)0, c01, false, false);
      c10 = __builtin_amdgcn_wmma_f32_16x16x4_f32(false, a1, false, b0, (short)0, c10, false, false);
      c11 = __builtin_amdgcn_wmma_f32_16x16x4_f32(false, a1, false, b1, (short)0, c11, false, false);
    }
    __syncthreads();   // all waves done with buffer bsel before it is refilled
  }

  // epilogue
  const int tileM = blockM + wr;
  const int tileN = blockN + wc;
  const int rowHi = (lane >> 4) << 3;   // +8 for upper lane half
#pragma unroll
  for (int r = 0; r < 8; ++r) {
    int gm0 = tileM + r + rowHi;
    int gm1 = gm0 + 16;
    int gn0 = tileN + l16;
    int gn1 = gn0 + 16;
    float v00 = c00[r] + bias[gn0];
    float v01 = c01[r] + bias[gn1];
    float v10 = c10[r] + bias[gn0];
    float v11 = c11[r] + bias[gn1];
    if (relu) {
      v00 = fmaxf(v00, 0.f); v01 = fmaxf(v01, 0.f);
      v10 = fmaxf(v10, 0.f); v11 = fmaxf(v11, 0.f);
    }
    if (residual) {
      v00 += residual[(size_t)gm0 * N + gn0];
      v01 += residual[(size_t)gm0 * N + gn1];
      v10 += residual[(size_t)gm1 * N + gn0];
      v11 += residual[(size_t)gm1 * N + gn1];
    }
    out[(size_t)gm0 * N + gn0] = v00;
    out[(size_t)gm0 * N + gn1] = v01;
    out[(size_t)gm1 * N + gn0] = v10;
    out[(size_t)gm1 * N + gn1] = v11;
  }
}

// ---------------------------------------------------------------------------
// Local 7x7 attention. One thread per (pixel, head); 8 head-threads per pixel
// are consecutive lanes, so attn_map head-reduction uses shfl_xor width=8.
// qkv rows: [q(0..255) | k(256..511) | v(512..767)], channel = head*32+d.
// Zero-padded taps contribute logit 0 (matching F.unfold zero padding).
// ---------------------------------------------------------------------------
__global__ __launch_bounds__(256) void local_attn_kernel(
    const float* __restrict__ qkv, float* __restrict__ attn_out,
    float* __restrict__ attn_map) {
  const int t    = blockIdx.x * 256 + threadIdx.x;  // 0..65535
  const int head = t & 7;
  const int m    = t >> 3;
  const int n    = m >> 12;
  const int p    = m & 4095;
  const int py   = p >> 6;
  const int px   = p & 63;
  const float scale = 0.1767766952966369f;  // 1/sqrt(32)

  const float* qp = qkv + (size_t)m * 768 + head * HDIM;
  float q[HDIM];
#pragma unroll
  for (int d4 = 0; d4 < 8; ++d4) {
    float4 v = *(const float4*)(qp + d4 * 4);
    q[d4 * 4 + 0] = v.x * scale;
    q[d4 * 4 + 1] = v.y * scale;
    q[d4 * 4 + 2] = v.z * scale;
    q[d4 * 4 + 3] = v.w * scale;
  }

  float wts[KS2];
#pragma unroll
  for (int kk = 0; kk < KS2; ++kk) {
    int yy = py + (kk / 7) - 3;
    int xx = px + (kk % 7) - 3;
    float acc = 0.f;
    if ((unsigned)yy < 64u && (unsigned)xx < 64u) {
      const float* kp = qkv + (size_t)((n << 12) + (yy << 6) + xx) * 768 + 256 + head * HDIM;
#pragma unroll
      for (int d4 = 0; d4 < 8; ++d4) {
        float4 kv = *(const float4*)(kp + d4 * 4);
        acc += q[d4 * 4 + 0] * kv.x + q[d4 * 4 + 1] * kv.y +
               q[d4 * 4 + 2] * kv.z + q[d4 * 4 + 3] * kv.w;
      }
    }
    wts[kk] = acc;   // zero-padded tap => logit 0, participates in softmax
  }

  // softmax over 49 taps
  float mx = wts[0];
#pragma unroll
  for (int kk = 1; kk < KS2; ++kk) mx = fmaxf(mx, wts[kk]);
  float s = 0.f;
#pragma unroll
  for (int kk = 0; kk < KS2; ++kk) { wts[kk] = expf(wts[kk] - mx); s += wts[kk]; }
  const float inv = 1.f / s;
#pragma unroll
  for (int kk = 0; kk < KS2; ++kk) wts[kk] *= inv;

  // weighted V accumulation
  float o[HDIM];
#pragma unroll
  for (int d = 0; d < HDIM; ++d) o[d] = 0.f;
#pragma unroll
  for (int kk = 0; kk < KS2; ++kk) {
    int yy = py + (kk / 7) - 3;
    int xx = px + (kk % 7) - 3;
    if ((unsigned)yy < 64u && (unsigned)xx < 64u) {
      const float w = wts[kk];
      const float* vp = qkv + (size_t)((n << 12) + (yy << 6) + xx) * 768 + 512 + head * HDIM;
#pragma unroll
      for (int d4 = 0; d4 < 8; ++d4) {
        float4 vv = *(const float4*)(vp + d4 * 4);
        o[d4 * 4 + 0] += w * vv.x;
        o[d4 * 4 + 1] += w * vv.y;
        o[d4 * 4 + 2] += w * vv.z;
        o[d4 * 4 + 3] += w * vv.w;
      }
    }
  }
  float* op = attn_out + (size_t)m * CCH + head * HDIM;
#pragma unroll
  for (int d4 = 0; d4 < 8; ++d4) {
    float4 vv = { o[d4 * 4 + 0], o[d4 * 4 + 1], o[d4 * 4 + 2], o[d4 * 4 + 3] };
    *(float4*)(op + d4 * 4) = vv;
  }

  // attn_map = mean over heads of softmax weights  -> [n, 49, h, w]
#pragma unroll
  for (int kk = 0; kk < KS2; ++kk) {
    float v = wts[kk];
    v += __shfl_xor(v, 1, 8);
    v += __shfl_xor(v, 2, 8);
    v += __shfl_xor(v, 4, 8);
    if (head == 0)
      attn_map[((size_t)n * KS2 + kk) * 4096 + p] = v * 0.125f;
  }
}

// ---------------------------------------------------------------------------
// BatchNorm (training stats, biased variance): one block per channel.
// stats[c] = mean, stats[256+c] = rsqrt(var + eps)
// ---------------------------------------------------------------------------
__global__ __launch_bounds__(256) void bn_stats_kernel(const float* __restrict__ y,
                                                       float* __restrict__ stats) {
  const int c = blockIdx.x;
  const int t = threadIdx.x;
  float s = 0.f, sq = 0.f;
  for (int i = t; i < MROWS; i += 256) {
    float v = y[(size_t)i * CCH + c];
    s += v; sq += v * v;
  }
  __shared__ float ls[256], lq[256];
  ls[t] = s; lq[t] = sq;
  __syncthreads();
  for (int off = 128; off > 0; off >>= 1) {
    if (t < off) { ls[t] += ls[t + off]; lq[t] += lq[t + off]; }
    __syncthreads();
  }
  if (t == 0) {
    float mean = ls[0] * (1.f / MROWS);
    float var  = lq[0] * (1.f / MROWS) - mean * mean;
    stats[c] = mean;
    stats[CCH + c] = rsqrtf(var + 1e-5f);
  }
}

__global__ __launch_bounds__(256) void bn_apply_kernel(
    const float* __restrict__ y, const float* __restrict__ stats,
    const float* __restrict__ g, const float* __restrict__ b,
    float* __restrict__ out) {
  int idx = blockIdx.x * 256 + threadIdx.x;   // M*C
  int c = idx & 255;
  out[idx] = (y[idx] - stats[c]) * stats[CCH + c] * g[c] + b[c];
}

// ---------------------------------------------------------------------------
// Host orchestration
// ---------------------------------------------------------------------------
extern "C" void kernel_launch(void* const* d_in, const int* in_sizes, int n_in,
                              void* d_out, int out_size, void* d_ws, size_t ws_size,
                              hipStream_t stream) {
  const float* feature = (const float*)d_in[0];
  const float* in_w    = (const float*)d_in[1];
  const float* in_b    = (const float*)d_in[2];
  const float* out_w   = (const float*)d_in[3];
  const float* out_b   = (const float*)d_in[4];
  const float* bn1_g   = (const float*)d_in[5];
  const float* bn1_b   = (const float*)d_in[6];
  const float* bn2_g   = (const float*)d_in[7];
  const float* bn2_b   = (const float*)d_in[8];
  const float* l1_w    = (const float*)d_in[9];
  const float* l1_b    = (const float*)d_in[10];
  const float* l2_w    = (const float*)d_in[11];
  const float* l2_b    = (const float*)d_in[12];

  float* out = (float*)d_out;                    // [2,256,64,64] then [2,49,64,64]
  float* ws  = (float*)d_ws;

  const size_t MC = (size_t)MROWS * CCH;         // 2,097,152
  float* buf_x    = ws;                          // layer input, NHWC [M,256]
  float* buf_qkv  = ws + MC;                     // [M,768]
  float* buf_attn = ws + MC + MROWS * 768;       // [M,256]
  float* buf_y    = buf_attn + MC;               // [M,256] pre-BN sum
  float* buf_x1   = buf_y + MC;                  // [M,256] BN1 output
  float* buf_ff   = buf_qkv;                     // [M,1024] aliases dead qkv+attn
  float* stats    = buf_x1 + MC;                 // [512]
  float* attn_map = out + MC;                    // [2,49,64,64]

  nchw_to_nhwc_kernel<<<MC / 256, 256, 0, stream>>>(feature, buf_x);

  for (int l = 0; l < 2; ++l) {
    const float* iw = in_w  + (size_t)l * 768 * 256;
    const float* ib = in_b  + (size_t)l * 768;
    const float* ow = out_w + (size_t)l * 256 * 256;
    const float* ob = out_b + (size_t)l * 256;
    const float* w1 = l1_w  + (size_t)l * 1024 * 256;
    const float* b1 = l1_b  + (size_t)l * 1024;
    const float* w2 = l2_w  + (size_t)l * 256 * 1024;
    const float* b2 = l2_b  + (size_t)l * 256;

    // qkv = x @ in_w^T + in_b          [8192 x 768], K=256
    wmma_gemm_kernel<<<dim3(768 / 128, MROWS / 64), 256, 0, stream>>>(
        buf_x, iw, ib, nullptr, buf_qkv, MROWS, 768, 256, 0);

    // local attention + attn_map (last layer's map survives)
    local_attn_kernel<<<(MROWS * NHEAD) / 256, 256, 0, stream>>>(
        buf_qkv, buf_attn, attn_map);

    // y = x + attn @ out_w^T + out_b   [8192 x 256], K=256
    wmma_gemm_kernel<<<dim3(256 / 128, MROWS / 64), 256, 0, stream>>>(
        buf_attn, ow, ob, buf_x, buf_y, MROWS, 256, 256, 0);

    // BN1 -> x1
    bn_stats_kernel<<<CCH, 256, 0, stream>>>(buf_y, stats);
    bn_apply_kernel<<<MC / 256, 256, 0, stream>>>(
        buf_y, stats, bn1_g + l * 256, bn1_b + l * 256, buf_x1);

    // ff = relu(x1 @ l1_w^T + l1_b)    [8192 x 1024], K=256
    wmma_gemm_kernel<<<dim3(1024 / 128, MROWS / 64), 256, 0, stream>>>(
        buf_x1, w1, b1, nullptr, buf_ff, MROWS, 1024, 256, 1);

    // y = x1 + ff @ l2_w^T + l2_b      [8192 x 256], K=1024
    wmma_gemm_kernel<<<dim3(256 / 128, MROWS / 64), 256, 0, stream>>>(
        buf_ff, w2, b2, buf_x1, buf_y, MROWS, 256, 1024, 0);

    // BN2 -> x (next layer input)
    bn_stats_kernel<<<CCH, 256, 0, stream>>>(buf_y, stats);
    bn_apply_kernel<<<MC / 256, 256, 0, stream>>>(
        buf_y, stats, bn2_g + l * 256, bn2_b + l * 256, buf_x);
  }

  nhwc_to_nchw_kernel<<<MC / 256, 256, 0, stream>>>(buf_x, out);
}